// GCNSingleHead_7164005450395
// MI455X (gfx1250) — compile-verified
//
#include <hip/hip_runtime.h>

// ---------------------------------------------------------------------------
// GCN single head on MI455X (gfx1250, wave32):
//   h   = (X @ W^T) * norm          -> f32 WMMA (V_WMMA_F32_16X16X4_F32)
//   agg = segment_sum(h[src], dst)  -> global_atomic_add_f32 (L2-resident)
//   out = relu(agg * norm)          -> elementwise
// ---------------------------------------------------------------------------

typedef float v2f __attribute__((ext_vector_type(2)));
typedef float v8f __attribute__((ext_vector_type(8)));

#define IN_FEATS   128
#define OUT_FEATS  64
#define LDS_STRIDE 132   // 128 + 4 pad: bank = (4*m + k) % 64 -> conflict-free b64

// ---------------------------------------------------------------------------
// Kernel 1: h[row][0:64] = (X[row] . W^T) * norm[row]
// Block = 128 threads = 4 waves; each wave owns a 16-row stripe, all 64 cols.
// A fragment (16x4 f32): lane 0-15 -> M=lane, K={k0,k0+1}; lane 16-31 -> M=lane-16, K={k0+2,k0+3}
// B fragment (4x16 f32): mirrored (lane&15 = N col, lane>>4 selects K pair) with B = W^T.
// ---------------------------------------------------------------------------
__global__ __launch_bounds__(128) void gcn_gemm_norm(
    const float* __restrict__ feature, const float* __restrict__ norm,
    const float* __restrict__ weight,  float* __restrict__ h, int nNodes)
{
    __shared__ float sW[OUT_FEATS * LDS_STRIDE];

    const int tid = threadIdx.x;
    // Stage W[64][128] into LDS with padded stride (coalesced float4 global reads).
    for (int i = tid; i < OUT_FEATS * (IN_FEATS / 4); i += 128) {
        const int r  = i >> 5;        // row (32 float4 per row)
        const int c4 = i & 31;
        float4 w4 = ((const float4*)weight)[i];
        float* p = &sW[r * LDS_STRIDE + c4 * 4];
        p[0] = w4.x; p[1] = w4.y; p[2] = w4.z; p[3] = w4.w;
    }
    __syncthreads();

    const int lane  = tid & 31;
    const int wave  = tid >> 5;
    const int m     = lane & 15;
    const int khalf = lane >> 4;      // 0 or 1
    const int kOff  = 2 * khalf;      // K sub-offset inside each 4-wide step

    const int rowBase = blockIdx.x * 64 + wave * 16;
    int rowA = rowBase + m;
    if (rowA >= nNodes) rowA = nNodes - 1;          // clamp loads; stores guarded
    const float* fRow = feature + (size_t)rowA * IN_FEATS;

    v8f acc0 = {}, acc1 = {}, acc2 = {}, acc3 = {};

    #pragma unroll 4
    for (int k0 = 0; k0 < IN_FEATS; k0 += 4) {
        v2f a  = *(const v2f*)(fRow + k0 + kOff);
        v2f b0 = *(const v2f*)(&sW[( 0 + m) * LDS_STRIDE + k0 + kOff]);
        v2f b1 = *(const v2f*)(&sW[(16 + m) * LDS_STRIDE + k0 + kOff]);
        v2f b2 = *(const v2f*)(&sW[(32 + m) * LDS_STRIDE + k0 + kOff]);
        v2f b3 = *(const v2f*)(&sW[(48 + m) * LDS_STRIDE + k0 + kOff]);
        acc0 = __builtin_amdgcn_wmma_f32_16x16x4_f32(false, a, false, b0, (short)0, acc0, false, false);
        acc1 = __builtin_amdgcn_wmma_f32_16x16x4_f32(false, a, false, b1, (short)0, acc1, false, false);
        acc2 = __builtin_amdgcn_wmma_f32_16x16x4_f32(false, a, false, b2, (short)0, acc2, false, false);
        acc3 = __builtin_amdgcn_wmma_f32_16x16x4_f32(false, a, false, b3, (short)0, acc3, false, false);
    }

    // C/D layout: VGPR r, lanes 0-15 -> M=r, N=lane; lanes 16-31 -> M=r+8, N=lane-16.
    float nrm[8];
    #pragma unroll
    for (int r = 0; r < 8; ++r) {
        int rr = rowBase + 8 * khalf + r;
        nrm[r] = norm[rr >= nNodes ? nNodes - 1 : rr];
    }
    #pragma unroll
    for (int r = 0; r < 8; ++r) {
        const int rr = rowBase + 8 * khalf + r;
        if (rr < nNodes) {
            float* hp = h + (size_t)rr * OUT_FEATS + m;
            hp[ 0] = acc0[r] * nrm[r];
            hp[16] = acc1[r] * nrm[r];
            hp[32] = acc2[r] * nrm[r];
            hp[48] = acc3[r] * nrm[r];
        }
    }
}

// ---------------------------------------------------------------------------
// Kernel 2: agg[dst] += h[src]  (16 lanes x float4 per edge; agg is L2-resident)
// ---------------------------------------------------------------------------
__global__ __launch_bounds__(256) void gcn_scatter(
    const float* __restrict__ h, const int* __restrict__ esrc,
    const int* __restrict__ edst, float* __restrict__ agg, int nEdges)
{
    const long tid = (long)blockIdx.x * 256 + threadIdx.x;
    const long e   = tid >> 4;
    if (e >= nEdges) return;
    const int c = (int)(tid & 15) * 4;

    const int s = esrc[e];
    const int d = edst[e];
    float4 v = *(const float4*)(h + (size_t)s * OUT_FEATS + c);
    float* p = agg + (size_t)d * OUT_FEATS + c;
    atomicAdd(p + 0, v.x);
    atomicAdd(p + 1, v.y);
    atomicAdd(p + 2, v.z);
    atomicAdd(p + 3, v.w);
}

// ---------------------------------------------------------------------------
// Kernel 3: out = relu(out * norm)  (in place; out was zeroed, then atomically built)
// ---------------------------------------------------------------------------
__global__ __launch_bounds__(256) void gcn_finalize(
    float* __restrict__ out, const float* __restrict__ norm, int n)
{
    const int i = blockIdx.x * 256 + threadIdx.x;
    if (i < n) {
        float v = out[i] * norm[i >> 6];   // OUT_FEATS == 64
        out[i] = v > 0.0f ? v : 0.0f;
    }
}

// ---------------------------------------------------------------------------
extern "C" void kernel_launch(void* const* d_in, const int* in_sizes, int n_in,
                              void* d_out, int out_size, void* d_ws, size_t ws_size,
                              hipStream_t stream)
{
    const float* feature = (const float*)d_in[0];   // [N, 128]
    const float* norm    = (const float*)d_in[1];   // [N, 1]
    const float* weight  = (const float*)d_in[2];   // [64, 128]
    const int*   esrc    = (const int*)d_in[3];     // [E]
    const int*   edst    = (const int*)d_in[4];     // [E]

    const int nNodes = in_sizes[1];
    const int nEdges = in_sizes[3];

    float* out = (float*)d_out;                     // [N, 64] accumulation target
    float* h   = (float*)d_ws;                      // [N, 64] scratch

    // Zero the aggregation buffer every call (graph-capture-safe).
    hipMemsetAsync(d_out, 0, (size_t)out_size * sizeof(float), stream);

    // 1) h = (X @ W^T) * norm  via f32 WMMA
    const int gemmBlocks = (nNodes + 63) / 64;
    gcn_gemm_norm<<<gemmBlocks, 128, 0, stream>>>(feature, norm, weight, h, nNodes);

    // 2) segment_sum via float atomics
    const long scatterThreads = (long)nEdges * 16;
    const int  scatterBlocks  = (int)((scatterThreads + 255) / 256);
    gcn_scatter<<<scatterBlocks, 256, 0, stream>>>(h, esrc, edst, out, nEdges);

    // 3) out = relu(out * norm)
    const int n = nNodes * OUT_FEATS;
    gcn_finalize<<<(n + 255) / 256, 256, 0, stream>>>(out, norm, n);
}